// OutputBlock_77524159693447
// MI455X (gfx1250) — compile-verified
//
#include <hip/hip_runtime.h>
#include <hip/hip_bf16.h>

typedef __attribute__((ext_vector_type(2))) float v2f;
typedef __attribute__((ext_vector_type(8))) float v8f;

#define HID   128
#define OE    256
#define OCH   128
#define NL    3
#define MT    16          // node rows per block tile
#define STR   260         // LDS row stride (floats): 260%64=4 -> conflict-free
#define EPB   16          // edges per block in scatter kernel

// ---------------------------------------------------------------- zero init
__global__ void zero_f32(float* __restrict__ p, long long n) {
    long long i = ((long long)blockIdx.x * blockDim.x + threadIdx.x) * 4;
    if (i + 3 < n) {
        *(float4*)(p + i) = make_float4(0.f, 0.f, 0.f, 0.f);
    } else {
        for (long long j = i; j < n; ++j) p[j] = 0.f;
    }
}

// ------------------------------------------- edge gating + scatter-add (atomics)
__global__ void __launch_bounds__(256)
edge_scatter(const float* __restrict__ x,        // [E,128]
             const float* __restrict__ rbf,      // [E,6]
             const long long* __restrict__ eidx, // [E]
             const float* __restrict__ W_rbf,    // [128,6]
             float* __restrict__ hn,             // [N,128] accumulator (d_out reused)
             int nE) {
    __shared__ float s_w[HID * 6];
    __shared__ float s_rbf[EPB * 6];
    __shared__ int   s_i[EPB];
    const int tid = threadIdx.x;
    const long long base = (long long)blockIdx.x * EPB;

    for (int t = tid; t < HID * 6; t += 256) s_w[t] = W_rbf[t];
    if (tid < EPB * 6) {
        long long off = base * 6 + tid;
        s_rbf[tid] = (off < (long long)nE * 6) ? rbf[off] : 0.f;
    }
    if (tid < EPB) {
        long long e = base + tid;
        s_i[tid] = (e < nE) ? (int)eidx[e] : 0;
    }
    __syncthreads();

    const int h   = tid & (HID - 1);
    const int sub = tid >> 7;                 // 0..1 (two edges in flight)
    const float* wr = &s_w[h * 6];
    for (int el = sub; el < EPB; el += 2) {
        long long e = base + el;
        if (e >= nE) break;
        const float* rb = &s_rbf[el * 6];
        float g = rb[0]*wr[0] + rb[1]*wr[1] + rb[2]*wr[2]
                + rb[3]*wr[3] + rb[4]*wr[4] + rb[5]*wr[5];
        float val = g * x[e * HID + h];
        unsafeAtomicAdd(&hn[(long long)s_i[el] * HID + h], val);  // global_atomic_add_f32
    }
}

// ------------------------------ fused node MLP: up-proj + 3 swish layers + out-proj
// Each wave owns TWO N-tiles (wave, wave+8) processed with dual independent
// WMMA accumulation chains sharing one A-fragment -> 2x load-level parallelism,
// half the LDS A traffic.
__global__ void __launch_bounds__(256)
node_mlp(const float* __restrict__ hn,    // [N,128]  (== d_out, consumed first)
         const float* __restrict__ W_up,  // [256,128]
         const float* __restrict__ Ws,    // [3,256,256]
         const float* __restrict__ bs,    // [3,256]
         const float* __restrict__ W_out, // [128,256]
         float* __restrict__ out,         // [N,128]  (== d_out)
         int nNodes) {
    __shared__ float sA[MT * STR];
    __shared__ float sB[MT * STR];
    const int tid  = threadIdx.x;
    const int lane = tid & 31;
    const int wave = tid >> 5;          // 0..7
    const int half = lane >> 4;         // 0/1
    const int l16  = lane & 15;
    const long long base = (long long)blockIdx.x * MT;

    // stage 0: load 16x128 input tile (coalesced) into sA
    #pragma unroll
    for (int r = 0; r < (MT * HID) / 256; ++r) {
        int t = r * 256 + tid;
        int m = t >> 7;
        int k = t & (HID - 1);
        long long row = base + m;
        float v = (row < nNodes) ? hn[row * HID + k] : 0.f;
        sA[m * STR + k] = v;
    }
    __syncthreads();

    // ---- up-projection: [16x128] @ W_up^T -> [16x256] in sB  (no bias)
    {
        const int n0 = wave * 16 + l16;          // tile nt = wave
        const int n1 = (wave + 8) * 16 + l16;    // tile nt = wave + 8
        v8f acc0 = {}, acc1 = {};
        #pragma unroll 8
        for (int k0 = 0; k0 < HID; k0 += 4) {
            const int kk = k0 + 2 * half;
            v2f a  = *(const v2f*)&sA[l16 * STR + kk];
            v2f b0 = *(const v2f*)&W_up[(long long)n0 * HID + kk];
            v2f b1 = *(const v2f*)&W_up[(long long)n1 * HID + kk];
            acc0 = __builtin_amdgcn_wmma_f32_16x16x4_f32(
                       false, a, false, b0, (short)0, acc0, false, false);
            acc1 = __builtin_amdgcn_wmma_f32_16x16x4_f32(
                       false, a, false, b1, (short)0, acc1, false, false);
        }
        #pragma unroll
        for (int v = 0; v < 8; ++v) {
            const int m = (v + 8 * half) * STR;
            sB[m + n0] = acc0[v];
            sB[m + n1] = acc1[v];
        }
    }
    __syncthreads();

    // ---- 3 hidden layers with swish, ping-pong sB -> sA -> sB -> sA
    float* src = sB;
    float* dst = sA;
    for (int l = 0; l < NL; ++l) {
        const float* W    = Ws + (long long)l * OE * OE;
        const float* bias = bs + l * OE;
        const int n0 = wave * 16 + l16;
        const int n1 = (wave + 8) * 16 + l16;
        v8f acc0 = {}, acc1 = {};
        #pragma unroll 8
        for (int k0 = 0; k0 < OE; k0 += 4) {
            const int kk = k0 + 2 * half;
            v2f a  = *(const v2f*)&src[l16 * STR + kk];
            v2f b0 = *(const v2f*)&W[(long long)n0 * OE + kk];
            v2f b1 = *(const v2f*)&W[(long long)n1 * OE + kk];
            acc0 = __builtin_amdgcn_wmma_f32_16x16x4_f32(
                       false, a, false, b0, (short)0, acc0, false, false);
            acc1 = __builtin_amdgcn_wmma_f32_16x16x4_f32(
                       false, a, false, b1, (short)0, acc1, false, false);
        }
        const float bv0 = bias[n0];
        const float bv1 = bias[n1];
        #pragma unroll
        for (int v = 0; v < 8; ++v) {
            const int m = (v + 8 * half) * STR;
            float t0 = acc0[v] + bv0;
            float t1 = acc1[v] + bv1;
            dst[m + n0] = t0 / (1.f + __expf(-t0));    // swish
            dst[m + n1] = t1 / (1.f + __expf(-t1));
        }
        __syncthreads();
        float* tmp = src; src = dst; dst = tmp;        // after 3 swaps: src == sA
    }

    // ---- out projection: [16x256] @ W_out^T -> [16x128] to global
    {
        const int n = wave * 16 + l16;       // 8 waves cover 128 cols
        v8f acc = {};
        #pragma unroll 8
        for (int k0 = 0; k0 < OE; k0 += 4) {
            const int kk = k0 + 2 * half;
            v2f a = *(const v2f*)&src[l16 * STR + kk];
            v2f b = *(const v2f*)&W_out[(long long)n * OE + kk];
            acc = __builtin_amdgcn_wmma_f32_16x16x4_f32(
                      false, a, false, b, (short)0, acc, false, false);
        }
        #pragma unroll
        for (int v = 0; v < 8; ++v) {
            long long row = base + v + 8 * half;
            if (row < nNodes) out[row * OCH + n] = acc[v];
        }
    }
}

// ---------------------------------------------------------------- launcher
extern "C" void kernel_launch(void* const* d_in, const int* in_sizes, int n_in,
                              void* d_out, int out_size, void* d_ws, size_t ws_size,
                              hipStream_t stream) {
    const float*     x     = (const float*)d_in[0];
    const float*     rbf   = (const float*)d_in[1];
    const long long* eidx  = (const long long*)d_in[2];
    // d_in[3] = num_nodes scalar (device); derive N from out_size instead
    const float*     W_rbf = (const float*)d_in[4];
    const float*     W_up  = (const float*)d_in[5];
    const float*     Ws    = (const float*)d_in[6];
    const float*     bsb   = (const float*)d_in[7];
    const float*     W_out = (const float*)d_in[8];
    float*           out   = (float*)d_out;

    const int E = in_sizes[0] / HID;
    const int N = out_size / OCH;

    // 1) zero the node accumulator (d_out doubles as h_nodes [N,128])
    long long total = (long long)N * HID;
    long long zb = (total + 256LL * 4 - 1) / (256LL * 4);
    zero_f32<<<(int)zb, 256, 0, stream>>>(out, total);

    // 2) edge gating + scatter-add into node accumulator
    int eb = (E + EPB - 1) / EPB;
    edge_scatter<<<eb, 256, 0, stream>>>(x, rbf, eidx, W_rbf, out, E);

    // 3) fused node-side MLP (reads its tile fully before overwriting it)
    int nb = (N + MT - 1) / MT;
    node_mlp<<<nb, 256, 0, stream>>>(out, W_up, Ws, bsb, W_out, out, N);
}